// ResGraphModule_1262720385735
// MI455X (gfx1250) — compile-verified
//
#include <hip/hip_runtime.h>
#include <hip/hip_bf16.h>

#define C128 128
#define EC 16
#define BN_EPS 1e-5f
#define ALPHA 0.5f
#define GRU_MS 2   // row-slabs per GRU block (B-fragment reuse factor)

typedef float v2f __attribute__((ext_vector_type(2)));
typedef float v8f __attribute__((ext_vector_type(8)));

__device__ __forceinline__ v8f wmma_f32(v2f a, v2f b, v8f c) {
    // V_WMMA_F32_16X16X4_F32 : D = A(16x4) * B(4x16) + C(16x16)
    return __builtin_amdgcn_wmma_f32_16x16x4_f32(
        false, a, false, b, (short)0, c, false, false);
}

__device__ __forceinline__ void atomic_add_f32(float* p, float v) {
    unsafeAtomicAdd(p, v);   // -> global_atomic_add_f32 on gfx1250
}

// ---------------------------------------------------------------- utilities
__global__ void zero_f32(float* __restrict__ p, long long n) {
    long long i = (long long)blockIdx.x * blockDim.x + threadIdx.x;
    long long stride = (long long)gridDim.x * blockDim.x;
    for (; i < n; i += stride) p[i] = 0.0f;
}

// ---------------------------------------------------------------- batchnorm
__global__ void bn_stats(const float* __restrict__ x, float* __restrict__ stats,
                         int n_nodes) {
    int c = threadIdx.x & (C128 - 1);
    int g = (int)((blockIdx.x * blockDim.x + threadIdx.x) >> 7);
    int ng = (int)((gridDim.x * blockDim.x) >> 7);
    float s = 0.0f, s2 = 0.0f;
    for (int r = g; r < n_nodes; r += ng) {
        float v = x[(size_t)r * C128 + c];
        s += v; s2 += v * v;
    }
    atomic_add_f32(&stats[c], s);
    atomic_add_f32(&stats[C128 + c], s2);
}

__global__ void bn_apply(const float* __restrict__ x, const float* __restrict__ stats,
                         const float* __restrict__ gamma, const float* __restrict__ beta,
                         float* __restrict__ x0, float* __restrict__ h, int n_nodes) {
    long long total = (long long)n_nodes * C128;
    long long i = (long long)blockIdx.x * blockDim.x + threadIdx.x;
    if (i >= total) return;
    int c = (int)(i & (C128 - 1));
    float inv_n = 1.0f / (float)n_nodes;
    float mean = stats[c] * inv_n;
    float var = stats[C128 + c] * inv_n - mean * mean;
    float v = (x[i] - mean) * rsqrtf(var + BN_EPS) * gamma[c] + beta[c];
    x0[i] = v;
    h[i] = v;
}

// ---------------------------------------------------------------- degree
__global__ void degree_kernel(const int* __restrict__ ei, float* __restrict__ deg,
                              int n_edges) {
    int i = blockIdx.x * blockDim.x + threadIdx.x;
    if (i < n_edges) {
        int dst = ei[n_edges + i];          // edge_index[1][i]
        atomic_add_f32(&deg[dst], 1.0f);
    }
}

__global__ void recip_deg(float* __restrict__ deg, int n_nodes) {
    int i = blockIdx.x * blockDim.x + threadIdx.x;
    if (i < n_nodes) deg[i] = 1.0f / fmaxf(deg[i], 1.0f);   // PyG mean clamp
}

// ---------------------------------------------------------------- conv GEMM
// m[N,128] = h[N,128] @ W[128,128].  256 persistent blocks, 8 waves each;
// wave w owns col tile w and holds its entire B panel (64 VGPRs) in registers
// across row slabs.
__global__ __launch_bounds__(256) void conv_gemm(
        const float* __restrict__ h, const float* __restrict__ W,
        float* __restrict__ m, int n_nodes, int n_slabs) {
    int lane = threadIdx.x & 31;
    int wave = threadIdx.x >> 5;
    int mr   = lane & 15;
    int koff = (lane >> 4) * 2;           // K sub-offset (0 or 2)
    int col  = wave * 16 + mr;

    v2f breg[32];
#pragma unroll
    for (int kk = 0; kk < 32; ++kk) {
        int k0 = kk * 4 + koff;
        v2f b;
        b.x = W[(size_t)k0 * C128 + col];
        b.y = W[(size_t)(k0 + 1) * C128 + col];
        breg[kk] = b;
    }

    int nmax = n_nodes - 1;
    for (int slab = blockIdx.x; slab < n_slabs; slab += gridDim.x) {
        int ar = slab * 16 + mr;
        if (ar > nmax) ar = nmax;
        v8f acc = {};
#pragma unroll
        for (int kk = 0; kk < 32; ++kk) {
            v2f a = *(const v2f*)(h + (size_t)ar * C128 + kk * 4 + koff);
            acc = wmma_f32(a, breg[kk], acc);
        }
        int rbase = slab * 16 + (lane >> 4) * 8;
        #pragma unroll
        for (int g = 0; g < 8; ++g) {
            int r = rbase + g;
            if (r < n_nodes) m[(size_t)r * C128 + col] = acc[g];
        }
    }
}

// ---------------------------------------------------------------- edge messages
// One wave per edge; lane handles 4 channels.  edge_W staged in LDS (float4,
// conflict-free).  msg = relu(m[src] + edge_attr[e] @ edge_W) * dinv[dst],
// scattered with hardware float atomics (mean-aggregation folded in).
__global__ __launch_bounds__(256) void edge_msg(
        const int* __restrict__ ei, const float* __restrict__ eattr,
        const float* __restrict__ eW, const float* __restrict__ m,
        const float* __restrict__ dinv, float* __restrict__ agg, int n_edges) {
    __shared__ float4 Ws[EC * 32];        // Ws[k*32 + j] = eW[k][4j..4j+3]
    for (int i = threadIdx.x; i < EC * 32; i += blockDim.x)
        Ws[i] = ((const float4*)eW)[i];
    __syncthreads();

    int lane = threadIdx.x & 31;
    int wid  = (int)(blockIdx.x * (blockDim.x >> 5) + (threadIdx.x >> 5));
    int nw   = (int)(gridDim.x * (blockDim.x >> 5));

    for (int e = wid; e < n_edges; e += nw) {
        int src = ei[e];
        int dst = ei[n_edges + e];
        float sc = dinv[dst];
        float4 acc = ((const float4*)(m + (size_t)src * C128))[lane];
        const float4* ea4 = (const float4*)(eattr + (size_t)e * EC);
        float4 a0 = ea4[0], a1 = ea4[1], a2 = ea4[2], a3 = ea4[3];
        float av[EC] = {a0.x,a0.y,a0.z,a0.w, a1.x,a1.y,a1.z,a1.w,
                        a2.x,a2.y,a2.z,a2.w, a3.x,a3.y,a3.z,a3.w};
#pragma unroll
        for (int k = 0; k < EC; ++k) {
            float4 wv = Ws[k * 32 + lane];
            acc.x = fmaf(av[k], wv.x, acc.x);
            acc.y = fmaf(av[k], wv.y, acc.y);
            acc.z = fmaf(av[k], wv.z, acc.z);
            acc.w = fmaf(av[k], wv.w, acc.w);
        }
        float* dp = agg + (size_t)dst * C128 + lane * 4;
        atomic_add_f32(dp + 0, fmaxf(acc.x, 0.0f) * sc);
        atomic_add_f32(dp + 1, fmaxf(acc.y, 0.0f) * sc);
        atomic_add_f32(dp + 2, fmaxf(acc.z, 0.0f) * sc);
        atomic_add_f32(dp + 3, fmaxf(acc.w, 0.0f) * sc);
    }
}

// ---------------------------------------------------------------- GRU cell
// gi = agg @ w_ih^T + b_ih ; gh = h @ w_hh^T + b_hh ; fused gates.
// LDS-free: each wave owns one 16-col gate slice and keeps all six
// accumulators (ir/iz/in/hr/hz/hn) in registers; the six C-tiles share the
// same (row,col) mapping so gates run directly on WMMA accumulators.
// GRU_MS row-slabs per block amortize the B-fragment (weight) loads.
__global__ __launch_bounds__(256) void gru_kernel(
        const float* __restrict__ agg, const float* __restrict__ h_in,
        const float* __restrict__ w_ih, const float* __restrict__ w_hh,
        const float* __restrict__ b_ih, const float* __restrict__ b_hh,
        float* __restrict__ h_out, int n_nodes) {
    int lane = threadIdx.x & 31;
    int wave = threadIdx.x >> 5;
    int mr   = lane & 15;
    int koff = (lane >> 4) * 2;
    int col  = wave * 16 + mr;            // hidden column 0..127
    int slab0 = blockIdx.x * GRU_MS;
    int nmax = n_nodes - 1;

    v8f air[GRU_MS], aiz[GRU_MS], ain[GRU_MS];
    v8f ahr[GRU_MS], ahz[GRU_MS], ahn[GRU_MS];
    v8f zed = {};
#pragma unroll
    for (int s = 0; s < GRU_MS; ++s) {
        air[s] = zed; aiz[s] = zed; ain[s] = zed;
        ahr[s] = zed; ahz[s] = zed; ahn[s] = zed;
    }
    int arow[GRU_MS];
#pragma unroll
    for (int s = 0; s < GRU_MS; ++s) {
        int r = (slab0 + s) * 16 + mr;
        arow[s] = (r > nmax) ? nmax : r;
    }

    for (int k = 0; k < C128; k += 4) {
        int k0 = k + koff;
        v2f bir = *(const v2f*)(w_ih + (size_t)col * C128 + k0);
        v2f biz = *(const v2f*)(w_ih + (size_t)(C128 + col) * C128 + k0);
        v2f bin = *(const v2f*)(w_ih + (size_t)(2 * C128 + col) * C128 + k0);
        v2f bhr = *(const v2f*)(w_hh + (size_t)col * C128 + k0);
        v2f bhz = *(const v2f*)(w_hh + (size_t)(C128 + col) * C128 + k0);
        v2f bhn = *(const v2f*)(w_hh + (size_t)(2 * C128 + col) * C128 + k0);
#pragma unroll
        for (int s = 0; s < GRU_MS; ++s) {
            v2f am = *(const v2f*)(agg  + (size_t)arow[s] * C128 + k0);
            v2f ah = *(const v2f*)(h_in + (size_t)arow[s] * C128 + k0);
            air[s] = wmma_f32(am, bir, air[s]);
            aiz[s] = wmma_f32(am, biz, aiz[s]);
            ain[s] = wmma_f32(am, bin, ain[s]);
            ahr[s] = wmma_f32(ah, bhr, ahr[s]);
            ahz[s] = wmma_f32(ah, bhz, ahz[s]);
            ahn[s] = wmma_f32(ah, bhn, ahn[s]);
        }
    }

    float br_i = b_ih[col], bz_i = b_ih[C128 + col], bn_i = b_ih[2 * C128 + col];
    float br_h = b_hh[col], bz_h = b_hh[C128 + col], bn_h = b_hh[2 * C128 + col];
#pragma unroll
    for (int s = 0; s < GRU_MS; ++s) {
        int rbase = (slab0 + s) * 16 + (lane >> 4) * 8;
#pragma unroll
        for (int g = 0; g < 8; ++g) {
            int r = rbase + g;
            if (r < n_nodes) {
                float ir  = air[s][g] + br_i;
                float iz  = aiz[s][g] + bz_i;
                float inn = ain[s][g] + bn_i;
                float hr  = ahr[s][g] + br_h;
                float hz  = ahz[s][g] + bz_h;
                float hn  = ahn[s][g] + bn_h;
                float rg = 1.0f / (1.0f + expf(-(ir + hr)));
                float zg = 1.0f / (1.0f + expf(-(iz + hz)));
                float ng = tanhf(inn + rg * hn);
                float hp = h_in[(size_t)r * C128 + col];
                h_out[(size_t)r * C128 + col] = (1.0f - zg) * ng + zg * hp;
            }
        }
    }
}

// ---------------------------------------------------------------- residual
__global__ void finalize(const float* __restrict__ h, float* __restrict__ out,
                         long long total) {
    long long i = (long long)blockIdx.x * blockDim.x + threadIdx.x;
    long long stride = (long long)gridDim.x * blockDim.x;
    for (; i < total; i += stride) {
        float x0 = out[i];                 // x0 stashed in d_out by bn_apply
        out[i] = (1.0f - ALPHA) * fmaxf(h[i], 0.0f) + ALPHA * x0;
    }
}

// ================================================================ host side
extern "C" void kernel_launch(void* const* d_in, const int* in_sizes, int n_in,
                              void* d_out, int out_size, void* d_ws, size_t ws_size,
                              hipStream_t stream) {
    const float* x        = (const float*)d_in[0];
    const int*   ei       = (const int*)  d_in[1];   // [2, E]
    const float* eattr    = (const float*)d_in[2];   // [E, 16]
    const float* bn_gamma = (const float*)d_in[3];
    const float* bn_beta  = (const float*)d_in[4];
    const float* edge_W   = (const float*)d_in[5];   // [16, 128]
    const float* conv_W   = (const float*)d_in[6];   // [2, 128, 128]
    const float* gru_w_ih = (const float*)d_in[7];   // [384, 128]
    const float* gru_w_hh = (const float*)d_in[8];   // [384, 128]
    const float* gru_b_ih = (const float*)d_in[9];   // [384]
    const float* gru_b_hh = (const float*)d_in[10];  // [384]
    float* out = (float*)d_out;

    const int n_nodes = in_sizes[0] / C128;
    const int n_edges = in_sizes[2] / EC;
    const long long nc = (long long)n_nodes * C128;

    // ---- workspace carving (256B aligned)
    char* w = (char*)d_ws;
    size_t off = 0;
    auto carve = [&](size_t bytes) {
        void* p = w + off;
        off = (off + bytes + 255) & ~(size_t)255;
        return p;
    };
    float* stats = (float*)carve(2 * C128 * sizeof(float));
    float* deg   = (float*)carve((size_t)n_nodes * sizeof(float));
    float* hA    = (float*)carve(nc * sizeof(float));
    float* hB    = (float*)carve(nc * sizeof(float));
    float* mbuf  = (float*)carve(nc * sizeof(float));
    float* agg   = (float*)carve(nc * sizeof(float));
    (void)ws_size; (void)n_in; (void)out_size;

    const int n_slabs = (n_nodes + 15) / 16;           // 3125 for N=50000

    // ---- BatchNorm (training-mode batch statistics)
    zero_f32<<<1, 256, 0, stream>>>(stats, 2 * C128);
    zero_f32<<<(n_nodes + 255) / 256, 256, 0, stream>>>(deg, n_nodes);
    bn_stats<<<256, 256, 0, stream>>>(x, stats, n_nodes);
    bn_apply<<<(int)((nc + 255) / 256), 256, 0, stream>>>(x, stats, bn_gamma, bn_beta,
                                                          out /*x0*/, hA, n_nodes);
    // ---- degree -> reciprocal (clamped)
    degree_kernel<<<(n_edges + 255) / 256, 256, 0, stream>>>(ei, deg, n_edges);
    recip_deg<<<(n_nodes + 255) / 256, 256, 0, stream>>>(deg, n_nodes);

    // ---- 2 GatedGraphConv layers
    float* h_cur = hA;
    float* h_nxt = hB;
    const int gru_blocks = (n_slabs + GRU_MS - 1) / GRU_MS;
    for (int layer = 0; layer < 2; ++layer) {
        conv_gemm<<<256, 256, 0, stream>>>(h_cur, conv_W + (size_t)layer * C128 * C128,
                                           mbuf, n_nodes, n_slabs);
        zero_f32<<<1024, 256, 0, stream>>>(agg, nc);
        edge_msg<<<2048, 256, 0, stream>>>(ei, eattr, edge_W, mbuf, deg, agg, n_edges);
        gru_kernel<<<gru_blocks, 256, 0, stream>>>(agg, h_cur, gru_w_ih, gru_w_hh,
                                                   gru_b_ih, gru_b_hh, h_nxt, n_nodes);
        float* t = h_cur; h_cur = h_nxt; h_nxt = t;
    }

    // ---- residual: out = 0.5*relu(h) + 0.5*x0 (x0 already in d_out)
    finalize<<<1024, 256, 0, stream>>>(h_cur, out, nc);
}